// RadialBasisArbitraryLayerT_77386720740135
// MI455X (gfx1250) — compile-verified
//
#include <hip/hip_runtime.h>

#define HPIX 65536   // 256*256 pixels
#define BB   8
#define KK   1024
#define TPB  256

// LDS layout: cpoint_loc [B][K][2] floats (16384) followed by alpha [B][K][2] (16384)
// = 131072 bytes dynamic LDS; 2 workgroups fit per 320KB WGP.

typedef int v4i __attribute__((vector_size(16)));
typedef __attribute__((address_space(1))) v4i* gv4i_p;   // global int4*
typedef __attribute__((address_space(3))) v4i* lv4i_p;   // LDS int4*

extern "C" __global__ void __launch_bounds__(TPB)
rbf_flow_kernel(const float* __restrict__ cpoint_loc,
                const float* __restrict__ alpha,
                const int*   __restrict__ sidx,
                const float* __restrict__ phi0,
                const float* __restrict__ phix,
                const float* __restrict__ phiy,
                const float* __restrict__ cp0,
                float*       __restrict__ out,
                int M)
{
    extern __shared__ float lds[];
    float* lds_loc = lds;                 // 16384 floats
    float* lds_alp = lds + BB * KK * 2;   // 16384 floats

    const int tid = threadIdx.x;

    // ---- Stage gather tables into LDS via CDNA5 async global->LDS engine ----
    // 4096 float4 per table, 256 threads -> exactly 16 iterations, no guard.
#if __has_builtin(__builtin_amdgcn_global_load_async_to_lds_b128)
#pragma unroll
    for (int j = 0; j < 16; ++j) {
        const int i = tid + j * TPB;
        __builtin_amdgcn_global_load_async_to_lds_b128(
            (gv4i_p)(cpoint_loc + i * 4), (lv4i_p)(lds_loc + i * 4), 0, 0);
        __builtin_amdgcn_global_load_async_to_lds_b128(
            (gv4i_p)(alpha + i * 4),      (lv4i_p)(lds_alp + i * 4), 0, 0);
    }
  #if __has_builtin(__builtin_amdgcn_s_wait_asynccnt)
    __builtin_amdgcn_s_wait_asynccnt(0);
  #else
    asm volatile("s_wait_asynccnt 0" ::: "memory");
  #endif
#else
#pragma unroll
    for (int j = 0; j < 16; ++j) {
        const int i = tid + j * TPB;
        ((float4*)lds_loc)[i] = ((const float4*)cpoint_loc)[i];
        ((float4*)lds_alp)[i] = ((const float4*)alpha)[i];
    }
#endif
    __syncthreads();

    // ---- One pixel per thread; all 8 batches accumulated in registers ----
    const int p = blockIdx.x * TPB + tid;   // flat pixel id (h*W + w)

    float accx[BB], accy[BB];
#pragma unroll
    for (int b = 0; b < BB; ++b) { accx[b] = 0.0f; accy[b] = 0.0f; }

    const float2* __restrict__ cp0v = (const float2*)cp0;
    const float2* __restrict__ locv = (const float2*)lds_loc;
    const float2* __restrict__ alpv = (const float2*)lds_alp;

    const long base = (long)p * (long)M;
    for (int m = 0; m < M; ++m) {
        const long e  = base + m;
        const int  k  = sidx[e];
        const float p0 = phi0[e];
        const float px = phix[e];
        const float py = phiy[e];
        const float2 c0 = cp0v[e];
        // phi = phi0 + (lx-cx0)*px + (ly-cy0)*py = psi + lx*px + ly*py
        const float psi = p0 - c0.x * px - c0.y * py;
#pragma unroll
        for (int b = 0; b < BB; ++b) {
            const float2 l = locv[b * KK + k];  // ds_load_b64 gather
            const float2 a = alpv[b * KK + k];  // ds_load_b64 gather
            const float phi = fmaf(l.x, px, fmaf(l.y, py, psi));
            accx[b] = fmaf(phi, a.x, accx[b]);
            accy[b] = fmaf(phi, a.y, accy[b]);
        }
    }

    // out[b][c][h][w], coalesced per plane
#pragma unroll
    for (int b = 0; b < BB; ++b) {
        out[(b * 2 + 0) * HPIX + p] = accx[b];
        out[(b * 2 + 1) * HPIX + p] = accy[b];
    }
}

extern "C" void kernel_launch(void* const* d_in, const int* in_sizes, int n_in,
                              void* d_out, int out_size, void* d_ws, size_t ws_size,
                              hipStream_t stream) {
    const float* cpl  = (const float*)d_in[0];  // cpoint_loc [B,K,2]
    const float* alp  = (const float*)d_in[1];  // alpha      [B,K,2]
    const int*   sidx = (const int*)  d_in[2];  // select_index [H,W,M]
    const float* p0   = (const float*)d_in[3];  // phi_0 [H,W,M]
    const float* px   = (const float*)d_in[4];  // phi_x [H,W,M]
    const float* py   = (const float*)d_in[5];  // phi_y [H,W,M]
    const float* c0   = (const float*)d_in[6];  // cpoints_0 [H,W,M,2]

    const int M = in_sizes[2] / HPIX;
    const size_t shmem = (size_t)(2 * BB * KK * 2) * sizeof(float);  // 131072 B

    (void)hipFuncSetAttribute((const void*)rbf_flow_kernel,
                              hipFuncAttributeMaxDynamicSharedMemorySize,
                              (int)shmem);

    rbf_flow_kernel<<<HPIX / TPB, TPB, shmem, stream>>>(
        cpl, alp, sidx, p0, px, py, c0, (float*)d_out, M);
}